// HarmonicEvolutionLayer_81913616269362
// MI455X (gfx1250) — compile-verified
//
#include <hip/hip_runtime.h>

typedef __attribute__((ext_vector_type(2))) float v2f;
typedef __attribute__((ext_vector_type(8))) float v8f;

#define SEQ     4096
#define LOG2SEQ 12
#define DDIM    1024
#define NBATCH  4
#define PI_F    3.14159265358979323846f

// ---------------------------------------------------------------------------
// Tensor Data Mover support (gfx1250). Falls back to plain loads if the
// builtin is unavailable (also keeps the host-side compile pass legal).
// ---------------------------------------------------------------------------
#if defined(__has_builtin)
#if __has_builtin(__builtin_amdgcn_tensor_load_to_lds) && \
    __has_builtin(__builtin_amdgcn_tensor_store_from_lds) && \
    __has_builtin(__builtin_amdgcn_s_wait_tensorcnt)
#define HAVE_TDM 1
#endif
#endif

#ifdef HAVE_TDM
typedef __attribute__((ext_vector_type(4))) unsigned int v4u;
typedef __attribute__((ext_vector_type(8))) int v8i;
typedef __attribute__((ext_vector_type(4))) int v4i;

// One 2D TDM tile: 4096 rows x 8 bytes, row stride 4096B (= D*4B), moved
// between global memory and a contiguous 32KB LDS region.
__device__ __forceinline__ void tdm_tile_col(bool is_store, const void* gptr,
                                             unsigned lds_off)
{
    const unsigned long long ga = (unsigned long long)(uintptr_t)gptr;
    // D# group 0: count=1 | lds_addr | global_addr[56:0] | type=2
    v4u g0;
    g0[0] = 1u;
    g0[1] = lds_off;
    g0[2] = (unsigned)ga;
    g0[3] = (unsigned)((ga >> 32) & 0x01FFFFFFull) | (2u << 30);
    // D# group 1: data_size=3 (8B); tensor_dim0=512 (8B units, = row length),
    // tensor_dim1=4096; tile_dim0=1, tile_dim1=4096, tile_dim2=0;
    // tensor_dim0_stride=512 (8B units = 4096B); tensor_dim1_stride=0.
    v8i g1;
    g1[0] = (int)(3u << 16);                        // mask=0 | data_size=3
    g1[1] = (int)(512u << 16);                      // tensor_dim0[15:0] @ bit48
    g1[2] = (int)((512u >> 16) | (4096u << 16));    // dim0[31:16] | dim1[15:0]
    g1[3] = (int)((4096u >> 16) | (1u << 16));      // dim1[31:16] | tile_dim0=1
    g1[4] = (int)(4096u);                           // tile_dim1 | tile_dim2=0
    g1[5] = (int)(512u);                            // stride0[31:0]
    g1[6] = 0;                                      // stride0[47:32] | stride1 lo
    g1[7] = 0;                                      // stride1 hi
    v4i gz = {0, 0, 0, 0};                          // groups 2/3 unused (2D)
#if __clang_major__ >= 23
    v8i gz8 = {0, 0, 0, 0, 0, 0, 0, 0};
    if (is_store) __builtin_amdgcn_tensor_store_from_lds(g0, g1, gz, gz, gz8, 0);
    else          __builtin_amdgcn_tensor_load_to_lds(g0, g1, gz, gz, gz8, 0);
#else
    if (is_store) __builtin_amdgcn_tensor_store_from_lds(g0, g1, gz, gz, 0);
    else          __builtin_amdgcn_tensor_load_to_lds(g0, g1, gz, gz, 0);
#endif
}
#endif // HAVE_TDM

// ---------------------------------------------------------------------------
// Kernel 1: circular filtering along S via in-LDS radix-2 FFT.
// One block handles one (batch, feature-pair). Two real sequences are packed
// into one complex FFT; the effective spectral filter is the symmetrized real
// kernel Ks[k] = (K[k]+K[N-k])/2 (the antisymmetric part is discarded by
// .real), so the packing needs no Hermitian separation. Forward DIF
// (natural -> bit-reversed), scale bins in bit-reversed order, inverse DIT
// (bit-reversed -> natural): no bit-reversal pass. The strided 4KB-pitch
// global tile is moved by the Tensor Data Mover in a single descriptor op.
// ---------------------------------------------------------------------------
__global__ __launch_bounds__(256)
void fft_filter_kernel(const float* __restrict__ x,
                       const float* __restrict__ freq_kernel,
                       float* __restrict__ out)
{
    __shared__ __align__(16) float2 z[SEQ];   // interleaved complex, 32KB

    const int tid = threadIdx.x;
    const int blk = blockIdx.x;              // [0, B*D/2)
    const int b   = blk / (DDIM / 2);
    const int d0  = (blk % (DDIM / 2)) * 2;
    const float* gbase = x + ((size_t)b * SEQ) * DDIM + d0;

#ifdef HAVE_TDM
    if ((tid >> 5) == 0) {                    // wave 0 drives the TDM
        tdm_tile_col(false, gbase, (unsigned)(uintptr_t)(void*)z);
        __builtin_amdgcn_s_wait_tensorcnt(0);
    }
    __syncthreads();
#else
    {
        const float2* xp = (const float2*)gbase;
        for (int s = tid; s < SEQ; s += 256)
            z[s] = xp[(size_t)s * (DDIM / 2)];
    }
    __syncthreads();
#endif

    // Forward radix-2 DIF, twiddle exp(-i*2*pi*k/m)
    for (int lh = LOG2SEQ - 1; lh >= 0; --lh) {
        const int   half = 1 << lh;
        const float step = -PI_F / (float)half;
        for (int t = tid; t < SEQ / 2; t += 256) {
            const int k  = t & (half - 1);
            const int i0 = ((t >> lh) << (lh + 1)) + k;
            const int i1 = i0 + half;
            float2 a = z[i0];
            float2 bq = z[i1];
            float2 sum; sum.x = a.x + bq.x; sum.y = a.y + bq.y;
            float  dr = a.x - bq.x, di = a.y - bq.y;
            float  sn, cs;
            __sincosf(step * (float)k, &sn, &cs);
            float2 tw; tw.x = dr * cs - di * sn; tw.y = dr * sn + di * cs;
            z[i0] = sum;
            z[i1] = tw;
        }
        __syncthreads();
    }

    // Pointwise: multiply bin (at bit-reversed position) by Ks[k]/N
    for (int p = tid; p < SEQ; p += 256) {
        unsigned k  = __brev((unsigned)p) >> (32 - LOG2SEQ);
        float    ks = 0.5f * (freq_kernel[k] + freq_kernel[(SEQ - k) & (SEQ - 1)])
                      * (1.0f / (float)SEQ);
        float2 v = z[p];
        v.x *= ks; v.y *= ks;
        z[p] = v;
    }
    __syncthreads();

    // Inverse radix-2 DIT, twiddle exp(+i*2*pi*k/m)
    for (int lh = 0; lh < LOG2SEQ; ++lh) {
        const int   half = 1 << lh;
        const float step = PI_F / (float)half;
        for (int t = tid; t < SEQ / 2; t += 256) {
            const int k  = t & (half - 1);
            const int i0 = ((t >> lh) << (lh + 1)) + k;
            const int i1 = i0 + half;
            float2 a = z[i0];
            float2 bq = z[i1];
            float  sn, cs;
            __sincosf(step * (float)k, &sn, &cs);
            float  tr = bq.x * cs - bq.y * sn;
            float  ti = bq.x * sn + bq.y * cs;
            float2 lo; lo.x = a.x + tr; lo.y = a.y + ti;
            float2 hi; hi.x = a.x - tr; hi.y = a.y - ti;
            z[i0] = lo;
            z[i1] = hi;
        }
        __syncthreads();
    }

    float* obase = out + ((size_t)b * SEQ) * DDIM + d0;
#ifdef HAVE_TDM
    if ((tid >> 5) == 0) {
        tdm_tile_col(true, obase, (unsigned)(uintptr_t)(void*)z);
        __builtin_amdgcn_s_wait_tensorcnt(0);  // keep LDS alive until drained
    }
#else
    {
        float2* op = (float2*)obase;
        for (int s = tid; s < SEQ; s += 256)
            op[(size_t)s * (DDIM / 2)] = z[s];
    }
#endif
}

// ---------------------------------------------------------------------------
// Kernel 2: quaternion rotation (einsum 'bstq,oq->bsto') on the matrix core
// via V_WMMA_F32_16X16X4_F32, then LayerNorm over the 1024-feature row.
// One block per (b,s) row; operates in place on `io`.
//   A (16x4, f32): lane<16 -> VGPR{0,1}=K{0,1}; lane>=16 -> K{2,3}; M = lane&15
//   B (4x16, f32): column n = lane&15; B[k][n] = rot[n][k] for n<4, else 0
//   D (16x16 f32): lane N=lane&15, VGPR v holds M = v + 8*(lane>>4)
// ---------------------------------------------------------------------------
__global__ __launch_bounds__(256)
void rotate_ln_kernel(float* __restrict__ io,
                      const float* __restrict__ rot,    // 4x4 row-major [o][q]
                      const float* __restrict__ gamma,
                      const float* __restrict__ beta)
{
    __shared__ __align__(16) float row[DDIM];
    __shared__ __align__(16) float rrow[DDIM];
    __shared__ float redS[8];
    __shared__ float redQ[8];
    __shared__ float s_mean, s_rstd;

    const int tid  = threadIdx.x;
    const int lane = tid & 31;
    const int wid  = tid >> 5;
    float* base = io + (size_t)blockIdx.x * DDIM;

    ((float4*)row)[tid] = ((const float4*)base)[tid];
    __syncthreads();

    const int m  = lane & 15;   // A: row M / B,D: column N
    const int kh = lane >> 4;   // selects K pair {0,1} vs {2,3}

    v2f bmat;
    bmat.x = (m < 4) ? rot[m * 4 + 2 * kh + 0] : 0.0f;
    bmat.y = (m < 4) ? rot[m * 4 + 2 * kh + 1] : 0.0f;

    #pragma unroll
    for (int iter = 0; iter < 2; ++iter) {
        const int qbase = wid * 32 + iter * 16;   // 16 quaternions per WMMA
        v2f a;
        a.x = row[(qbase + m) * 4 + 2 * kh + 0];
        a.y = row[(qbase + m) * 4 + 2 * kh + 1];
        v8f c = {0.f, 0.f, 0.f, 0.f, 0.f, 0.f, 0.f, 0.f};
        v8f d = __builtin_amdgcn_wmma_f32_16x16x4_f32(
            /*neg_a=*/false, a, /*neg_b=*/false, bmat,
            /*c_mod=*/(short)0, c, /*reuse_a=*/false, /*reuse_b=*/false);
        if (m < 4) {                 // only columns 0..3 carry the rotation
            #pragma unroll
            for (int vv = 0; vv < 8; ++vv) {
                const int qq = qbase + vv + 8 * kh;
                rrow[qq * 4 + m] = d[vv];
            }
        }
    }
    __syncthreads();

    // LayerNorm over the row (population variance, eps = 1e-5)
    float v0 = rrow[tid * 4 + 0];
    float v1 = rrow[tid * 4 + 1];
    float v2 = rrow[tid * 4 + 2];
    float v3 = rrow[tid * 4 + 3];
    float s = v0 + v1 + v2 + v3;
    float q = v0 * v0 + v1 * v1 + v2 * v2 + v3 * v3;
    #pragma unroll
    for (int off = 16; off > 0; off >>= 1) {
        s += __shfl_down(s, off);
        q += __shfl_down(q, off);
    }
    if (lane == 0) { redS[wid] = s; redQ[wid] = q; }
    __syncthreads();
    if (tid == 0) {
        float ts = 0.f, tq = 0.f;
        #pragma unroll
        for (int i = 0; i < 8; ++i) { ts += redS[i]; tq += redQ[i]; }
        float mean = ts * (1.0f / DDIM);
        float var  = tq * (1.0f / DDIM) - mean * mean;
        s_mean = mean;
        s_rstd = rsqrtf(var + 1e-5f);
    }
    __syncthreads();

    const float mean = s_mean, rstd = s_rstd;
    float4 g  = ((const float4*)gamma)[tid];
    float4 bt = ((const float4*)beta)[tid];
    float4 o;
    o.x = (v0 - mean) * rstd * g.x + bt.x;
    o.y = (v1 - mean) * rstd * g.y + bt.y;
    o.z = (v2 - mean) * rstd * g.z + bt.z;
    o.w = (v3 - mean) * rstd * g.w + bt.w;
    ((float4*)base)[tid] = o;
}

// ---------------------------------------------------------------------------
extern "C" void kernel_launch(void* const* d_in, const int* in_sizes, int n_in,
                              void* d_out, int out_size, void* d_ws, size_t ws_size,
                              hipStream_t stream)
{
    const float* x    = (const float*)d_in[0];  // (4,4096,1024)
    const float* rotm = (const float*)d_in[1];  // (4,4)
    const float* fk   = (const float*)d_in[2];  // (4096,)
    const float* g    = (const float*)d_in[3];  // (1024,)
    const float* be   = (const float*)d_in[4];  // (1024,)
    float* out = (float*)d_out;                 // (4,4096,1024)

    // Pass 1: spectral filter -> writes filtered tensor into d_out (TDM-fed)
    fft_filter_kernel<<<NBATCH * (DDIM / 2), 256, 0, stream>>>(x, fk, out);
    // Pass 2: quaternion rotation (WMMA) + LayerNorm, in place on d_out
    rotate_ln_kernel<<<NBATCH * SEQ, 256, 0, stream>>>(out, rotm, g, be);
}